// DGCLoss_66348654789251
// MI455X (gfx1250) — compile-verified
//
#include <hip/hip_runtime.h>
#include <math.h>

// ---------------------------------------------------------------------------
// DGCLoss for MI455X (gfx1250, wave32).
//   N=384 samples, D=256 dims, M=N-1.
//   Phase 1: row inverse norms.
//   Phase 2: S = X @ X^T via V_WMMA_F32_16X16X4_F32 (fp32 WMMA, precision
//            matters: K=0.001 sigmoid amplifies cosine error x1000).
//   Phase 3: per-row soft-rank sigmoid sum (N*M^2 ~ 56M exp's, VALU trans
//            bound), DCG + histogram-based ideal DCG (relevance has only
//            8 possible values (g clamped to 7), so "sort" == 8-bin prefix sum).
//   Phase 4: mean over valid rows -> loss = 1 - mean(ndcg).
// ---------------------------------------------------------------------------

typedef float v2f __attribute__((ext_vector_type(2)));
typedef float v8f __attribute__((ext_vector_type(8)));

#define NROW 384
#define DDIM 256
#define MM   (NROW - 1)   // 383

// ---------------------------------------------------------------- kernel 1 --
__global__ void dgc_inv_norm_kernel(const float* __restrict__ x,
                                    float* __restrict__ inv_norm) {
    __shared__ float red[DDIM];
    const int row = blockIdx.x;
    const int t   = threadIdx.x;            // 0..255
    const float v = x[row * DDIM + t];
    red[t] = v * v;
    __syncthreads();
    #pragma unroll
    for (int s = DDIM / 2; s > 0; s >>= 1) {
        if (t < s) red[t] += red[t + s];
        __syncthreads();
    }
    if (t == 0) {
        float nrm = fmaxf(sqrtf(red[0]), 1e-8f);   // clip(norm, EPS)
        inv_norm[row] = 1.0f / nrm;
    }
}

// ---------------------------------------------------------------- kernel 2 --
// One wave32 per 16x16 output tile. K loop in steps of 4 using fp32 WMMA.
// A-matrix 16x4 f32 layout: lane<16 holds (M=lane, K=k0+{0,1}),
//                           lane>=16 holds (M=lane-16, K=k0+{2,3}).
// B = X^T, so B tile loads are symmetric with A (B[k][n] = X[n][k]).
// C/D layout: VGPR v -> row (lane>=16 ? 8 : 0)+v, col lane&15.
__global__ void dgc_gemm_rank_kernel(const float* __restrict__ x,
                                     const float* __restrict__ inv_norm,
                                     float* __restrict__ rankmat) {
    const int tm   = blockIdx.x;            // 0..23
    const int tn   = blockIdx.y;            // 0..23
    const int lane = threadIdx.x;           // 0..31
    const int m16  = lane & 15;
    const int kh   = lane >> 4;             // 0 or 1

    const float* arow = x + (size_t)(tm * 16 + m16) * DDIM + 2 * kh;
    const float* brow = x + (size_t)(tn * 16 + m16) * DDIM + 2 * kh;

    v8f c = {};
    #pragma unroll 8
    for (int k0 = 0; k0 < DDIM; k0 += 4) {
        v2f a = *(const v2f*)(arow + k0);
        v2f b = *(const v2f*)(brow + k0);
        // (neg_a, A, neg_b, B, c_mod, C, reuse_a, reuse_b)
        c = __builtin_amdgcn_wmma_f32_16x16x4_f32(false, a, false, b,
                                                  (short)0, c, false, false);
    }

    const int   col = tn * 16 + m16;
    const float inc = inv_norm[col];
    #pragma unroll
    for (int v = 0; v < 8; ++v) {
        const int   row  = tm * 16 + kh * 8 + v;
        const float cosv = c[v] * inv_norm[row] * inc;
        rankmat[(size_t)row * NROW + col] = (cosv + 1.0f) * 0.5f;
    }
}

// ---------------------------------------------------------------- kernel 3 --
// One 384-thread block per row n. Thread i < M owns off-diagonal column
// col = i + (i>=n). Diagonal of the j-loop handled branchlessly: sigma(0)=0.5
// is included in the sum, and indicator = sum - 0.5 + 1 = sum + 0.5.
__global__ void dgc_row_ndcg_kernel(const float* __restrict__ rankmat,
                                    const int* __restrict__ gt,
                                    float* __restrict__ ndcg,
                                    float* __restrict__ validf) {
    __shared__ float r[MM];
    __shared__ int   hist[8];
    __shared__ float redA[512];
    __shared__ float redB[512];

    const int n = blockIdx.x;
    const int t = threadIdx.x;              // 0..383

    if (t < 8) hist[t] = 0;
    __syncthreads();

    const int gtn = gt[n];
    float myrel = 0.0f;
    if (t < MM) {
        const int col = t + (t >= n ? 1 : 0);
        r[t] = rankmat[(size_t)n * NROW + col];
        int g = gt[col] - gtn;
        g = g < 0 ? -g : g;
        if (g > 7) g = 7;                   // safety clamp
        myrel = exp2f(fmaxf(10.0f - (float)g, 0.0f)) - 1.0f;
        atomicAdd(&hist[g], 1);
    }
    __syncthreads();

    // ---- soft-rank indicator + DCG contribution -------------------------
    float dcg_i = 0.0f;
    if (t < MM) {
        const float ri = r[t];
        float s = 0.0f;
        for (int j = 0; j < MM; ++j) {      // LDS broadcast read, no conflicts
            // arg = clip(-dij/K) = clip((r_i - r_j)*1000, -50, 50)
            float arg = (ri - r[j]) * 1000.0f;
            arg = fminf(fmaxf(arg, -50.0f), 50.0f);
            const float e = __expf(arg);
            s += __builtin_amdgcn_rcpf(1.0f + e);
        }
        const float ind = s + 0.5f;         // remove sigma(0)=0.5, add +1
        dcg_i = myrel * __builtin_amdgcn_rcpf(__log2f(ind + 1.0f));
    }

    // ---- ideal DCG: relevance sorted descending == g ascending ----------
    float idcg_i = 0.0f;
    if (t < MM) {
        int acc = 0, bin = 7;
        #pragma unroll
        for (int b = 0; b < 8; ++b) {
            const int cb = hist[b];
            if (t >= acc && t < acc + cb) bin = b;
            acc += cb;
        }
        const float relSorted = exp2f(fmaxf(10.0f - (float)bin, 0.0f)) - 1.0f;
        idcg_i = relSorted * __builtin_amdgcn_rcpf(__log2f((float)t + 2.0f));
    }

    // ---- dual block reduction (pad 384 -> 512) --------------------------
    redA[t] = dcg_i;
    redB[t] = idcg_i;
    if (t < 128) { redA[t + 384] = 0.0f; redB[t + 384] = 0.0f; }
    __syncthreads();
    #pragma unroll
    for (int s = 256; s > 0; s >>= 1) {
        if (t < s) { redA[t] += redA[t + s]; redB[t] += redB[t + s]; }
        __syncthreads();
    }
    if (t == 0) {
        const float dcg  = redA[0];
        const float idcg = redB[0];
        const bool  valid = (idcg != 0.0f);
        ndcg[n]   = valid ? dcg / idcg : 0.0f;
        validf[n] = valid ? 1.0f : 0.0f;
    }
}

// ---------------------------------------------------------------- kernel 4 --
__global__ void dgc_finalize_kernel(const float* __restrict__ ndcg,
                                    const float* __restrict__ validf,
                                    float* __restrict__ out) {
    __shared__ float redA[512];
    __shared__ float redB[512];
    const int t = threadIdx.x;              // 0..383
    redA[t] = ndcg[t];
    redB[t] = validf[t];
    if (t < 128) { redA[t + 384] = 0.0f; redB[t + 384] = 0.0f; }
    __syncthreads();
    #pragma unroll
    for (int s = 256; s > 0; s >>= 1) {
        if (t < s) { redA[t] += redA[t + s]; redB[t] += redB[t + s]; }
        __syncthreads();
    }
    if (t == 0) {
        const float cnt  = redB[0];
        const float mean = redA[0] / fmaxf(cnt, 1.0f);
        out[0] = (cnt == 0.0f) ? 1.0f : (1.0f - mean);
    }
}

// ---------------------------------------------------------------------------
extern "C" void kernel_launch(void* const* d_in, const int* in_sizes, int n_in,
                              void* d_out, int out_size, void* d_ws, size_t ws_size,
                              hipStream_t stream) {
    const float* x  = (const float*)d_in[0];   // [384, 256] fp32
    const int*   gt = (const int*)d_in[1];     // [384] int32
    float* out = (float*)d_out;                // scalar loss

    float* ws       = (float*)d_ws;
    float* rankmat  = ws;                              // N*N floats
    float* inv_norm = rankmat + (size_t)NROW * NROW;   // N floats
    float* ndcg     = inv_norm + NROW;                 // N floats
    float* validf   = ndcg + NROW;                     // N floats

    dgc_inv_norm_kernel<<<NROW, DDIM, 0, stream>>>(x, inv_norm);
    dgc_gemm_rank_kernel<<<dim3(NROW / 16, NROW / 16), 32, 0, stream>>>(
        x, inv_norm, rankmat);
    dgc_row_ndcg_kernel<<<NROW, 384, 0, stream>>>(rankmat, gt, ndcg, validf);
    dgc_finalize_kernel<<<1, 384, 0, stream>>>(ndcg, validf, out);
}